// dice_cldice_loss_14465449853038
// MI455X (gfx1250) — compile-verified
//
#include <hip/hip_runtime.h>
#include <hip/hip_bf16.h>

// Problem shape (from reference): y_pred [2,2,96,192,192] f32, y_true [2,1,96,192,192] int
#define DD 96
#define HH 192
#define WW 192
#define HW (HH * WW)
#define VOL (DD * HH * WW)          // 3,538,944 per (b,c) volume

static constexpr int RED_BLOCKS = 1024;
// Workspace layout (floats): acc[16] | partials[RED_BLOCKS*3] | bufA[VOL] | bufB[VOL] | skel[VOL]
// Required ws bytes = (16 + 3072 + 3*VOL)*4 ~= 42.5 MB  (whole working set is L2-resident)

__device__ __forceinline__ int iclampi(int x, int lo, int hi) {
  return x < lo ? lo : (x > hi ? hi : x);
}

// softmax channel-1 of a 2-channel softmax: 1/(1+e^{x0-x1})
__device__ __forceinline__ float softmax1(float x0, float x1) {
  return 1.0f / (1.0f + __expf(x0 - x1));
}

// ---- CDNA5 async global->LDS DMA (ASYNCcnt-tracked) ----
__device__ __forceinline__ void async_load_f32(unsigned lds_byte_addr, const float* gsrc) {
  asm volatile("global_load_async_to_lds_b32 %0, %1, off"
               :: "v"(lds_byte_addr), "v"(gsrc)
               : "memory");
}
template <int N>
__device__ __forceinline__ void wait_async() {
  asm volatile("s_wait_asynccnt %0" :: "i"(N) : "memory");
}

// ----------------------------------------------------------------------
__global__ void k_zero_acc(float* acc) {
  if (threadIdx.x < 16) acc[threadIdx.x] = 0.0f;
}

// mode 0: out = softmax ch1 of y_pred[b]; mode 1: out = (y_true[b]==1)
__global__ __launch_bounds__(256) void k_init_field(float* __restrict__ out,
    const float* __restrict__ yp, const int* __restrict__ yt, int b, int mode) {
  int i = blockIdx.x * 256 + threadIdx.x;      // float4 index
  if (i >= VOL / 4) return;
  float4 r;
  if (mode == 0) {
    float4 x0 = reinterpret_cast<const float4*>(yp + (size_t)(2 * b) * VOL)[i];
    float4 x1 = reinterpret_cast<const float4*>(yp + (size_t)(2 * b + 1) * VOL)[i];
    r.x = softmax1(x0.x, x1.x); r.y = softmax1(x0.y, x1.y);
    r.z = softmax1(x0.z, x1.z); r.w = softmax1(x0.w, x1.w);
  } else {
    int4 t = reinterpret_cast<const int4*>(yt + (size_t)b * VOL)[i];
    r.x = (t.x == 1) ? 1.0f : 0.0f; r.y = (t.y == 1) ? 1.0f : 0.0f;
    r.z = (t.z == 1) ? 1.0f : 0.0f; r.w = (t.w == 1) ? 1.0f : 0.0f;
  }
  reinterpret_cast<float4*>(out)[i] = r;
}

// One fused skeleton stage:
//   e = erode7(prev)           (computed in LDS from async-DMA'd halo-2 planes)
//   cur = e                    (global store, skipped on last stage)
//   dil = dilate27(e)          (3x3 in-plane max in regs + 3-plane max)
//   delta = relu(prev - dil)   (prev center read from LDS raw ring)
//   k==0: skel = delta;  else: skel += relu(delta - skel*delta)
__global__ __launch_bounds__(256) void k_stage(
    const float* __restrict__ prev, float* __restrict__ cur,
    float* __restrict__ skel, int first, int write_cur) {
  constexpr int TW = 32, TH = 8, TD = 24;
  constexpr int RW2 = TW + 4, RH2 = TH + 4, SP2 = 40;  // raw halo-2 plane: 36x12, stride 40
  constexpr int RW1 = TW + 2, RH1 = TH + 2, SP1 = 36;  // eroded halo-1 plane: 34x10, stride 36
  __shared__ float raw[5][RH2 * SP2];                  // 5-slot ring (pipelined DMA)
  __shared__ float ebuf[3][RH1 * SP1];                 // 3-slot eroded-plane ring
  const int t = threadIdx.x;
  const int tx = t & 31, ty = t >> 5;
  const int w0 = blockIdx.x * TW;
  const int h0 = blockIdx.y * TH;
  const int d0 = blockIdx.z * TD;

  // DMA one halo-2 plane of `prev` (logical index p; contents = plane clamp(p)).
  // Exactly 2 async instructions per thread (dummy loads pad 432 -> 512) so the
  // per-wave ASYNCcnt arithmetic below (wait <= 2) is uniform across waves.
  auto load_plane = [&](int p) {
    const int slot = (p + 10) % 5;
    const float* plane = prev + (size_t)iclampi(p, 0, DD - 1) * HW;
#pragma unroll
    for (int rep = 0; rep < 2; ++rep) {
      int e = t + rep * 256;
      unsigned dst;
      const float* src;
      if (e < RH2 * RW2) {
        int r = e / RW2, c = e - r * RW2;
        int gh = iclampi(h0 + r - 2, 0, HH - 1);     // clamp == +/-inf pad for min/max
        int gw = iclampi(w0 + c - 2, 0, WW - 1);
        dst = (unsigned)(size_t)&raw[slot][r * SP2 + c];
        src = plane + gh * WW + gw;
      } else {                                        // dummy into pad cell (row 0, col 36)
        dst = (unsigned)(size_t)&raw[slot][RW2];
        src = plane;
      }
      async_load_f32(dst, src);
    }
  };

  // Erode logical plane q (clamped) into e-ring; optionally store interior to `cur`.
  auto erode_plane = [&](int q, bool store) {
    const int qc = iclampi(q, 0, DD - 1);
    const float* rM = raw[(qc - 1 + 10) % 5];        // contents raw(clamp(qc-1))
    const float* rC = raw[(qc     + 10) % 5];
    const float* rP = raw[(qc + 1 + 10) % 5];        // contents raw(clamp(qc+1))
    float* ep = ebuf[(q + 9) % 3];
#pragma unroll
    for (int rep = 0; rep < 2; ++rep) {
      int e = t + rep * 256;
      if (e < RH1 * RW1) {
        int r = e / RW1, c = e - r * RW1;            // e-plane coords
        int bix = (r + 1) * SP2 + (c + 1);           // raw-plane coords (halo-2)
        float m = rC[bix];
        m = fminf(m, rC[bix - 1]);   m = fminf(m, rC[bix + 1]);
        m = fminf(m, rC[bix - SP2]); m = fminf(m, rC[bix + SP2]);
        m = fminf(m, rM[bix]);       m = fminf(m, rP[bix]);
        ep[r * SP1 + c] = m;
        if (store && r >= 1 && r <= TH && c >= 1 && c <= TW) {
          cur[(size_t)qc * HW + (size_t)(h0 + r - 1) * WW + (size_t)(w0 + c - 1)] = m;
        }
      }
    }
  };

  // 3x3 in-plane max of eroded plane q at this thread's (ty,tx) output position.
  auto pmax2d = [&](int q) -> float {
    const float* r = &ebuf[(q + 9) % 3][ty * SP1 + tx];
    float m = r[0];
    m = fmaxf(m, r[1]);         m = fmaxf(m, r[2]);
    m = fmaxf(m, r[SP1]);       m = fmaxf(m, r[SP1 + 1]);     m = fmaxf(m, r[SP1 + 2]);
    m = fmaxf(m, r[2 * SP1]);   m = fmaxf(m, r[2 * SP1 + 1]); m = fmaxf(m, r[2 * SP1 + 2]);
    return m;
  };

  // Prologue: DMA planes d0-2 .. d0+2 (in order; last pair may remain in flight).
  load_plane(d0 - 2); load_plane(d0 - 1); load_plane(d0);
  load_plane(d0 + 1); load_plane(d0 + 2);
  wait_async<2>();                 // d0-2..d0+1 resident; d0+2 may be in flight
  __syncthreads();
  erode_plane(d0 - 1, false);      // halo plane (owned by neighbor block / clamp dup)
  erode_plane(d0, write_cur != 0); // this block owns plane d0's store
  __syncthreads();
  float pm_m = pmax2d(d0 - 1);
  float pm_c = pmax2d(d0);

  for (int d = d0; d < d0 + TD; ++d) {
    load_plane(d + 3);             // prefetch one plane ahead of need
    wait_async<2>();               // raw(d+2) resident; raw(d+3) still in flight
    __syncthreads();
    erode_plane(d + 1, (write_cur != 0) && (d + 1 <= d0 + TD - 1));
    __syncthreads();
    float pm_p = pmax2d(d + 1);
    float dil = fmaxf(pm_m, fmaxf(pm_c, pm_p));
    float center = raw[(d + 10) % 5][(ty + 2) * SP2 + (tx + 2)];  // prev[vid] from LDS
    float delta = fmaxf(center - dil, 0.0f);
    size_t vid = (size_t)d * HW + (size_t)(h0 + ty) * WW + (size_t)(w0 + tx);
    if (first) {
      skel[vid] = delta;
    } else {
      float s = skel[vid];
      skel[vid] = s + fmaxf(delta - s * delta, 0.0f);
    }
    pm_m = pm_c; pm_c = pm_p;
  }
}

// Dice sums over both batches: sum(oh1), sum(p1), sum(p1*oh1) -> partials[blk*3+j]
__global__ __launch_bounds__(256) void k_reduce_dice(
    const float* __restrict__ yp, const int* __restrict__ yt, float* __restrict__ partials) {
  float s_g = 0.0f, s_p = 0.0f, s_pg = 0.0f;
  const int t = threadIdx.x;
  for (int b = 0; b < 2; ++b) {
    const float* x0p = yp + (size_t)(2 * b) * VOL;
    const float* x1p = yp + (size_t)(2 * b + 1) * VOL;
    const int* tp = yt + (size_t)b * VOL;
    for (int v = blockIdx.x * 256 + t; v < VOL; v += RED_BLOCKS * 256) {
      float p = softmax1(x0p[v], x1p[v]);
      float g = (tp[v] == 1) ? 1.0f : 0.0f;
      s_g += g; s_p += p; s_pg += p * g;
    }
  }
  __shared__ float sm0[256], sm1[256], sm2[256];
  sm0[t] = s_g; sm1[t] = s_p; sm2[t] = s_pg;
  __syncthreads();
  for (int off = 128; off > 0; off >>= 1) {
    if (t < off) { sm0[t] += sm0[t + off]; sm1[t] += sm1[t + off]; sm2[t] += sm2[t + off]; }
    __syncthreads();
  }
  if (t == 0) {
    partials[blockIdx.x * 3 + 0] = sm0[0];
    partials[blockIdx.x * 3 + 1] = sm1[0];
    partials[blockIdx.x * 3 + 2] = sm2[0];
  }
}

// Skeleton sums for batch b: mode 0 -> weight = one-hot; mode 1 -> weight = p1 (recomputed)
__global__ __launch_bounds__(256) void k_reduce_skel(
    const float* __restrict__ skel, const float* __restrict__ yp, const int* __restrict__ yt,
    int b, int mode, float* __restrict__ partials) {
  float s_sw = 0.0f, s_s = 0.0f;
  const int t = threadIdx.x;
  const float* x0p = yp + (size_t)(2 * b) * VOL;
  const float* x1p = yp + (size_t)(2 * b + 1) * VOL;
  const int* tp = yt + (size_t)b * VOL;
  for (int v = blockIdx.x * 256 + t; v < VOL; v += RED_BLOCKS * 256) {
    float s = skel[v];
    float wgt = (mode == 0) ? ((tp[v] == 1) ? 1.0f : 0.0f)
                            : softmax1(x0p[v], x1p[v]);
    s_sw += s * wgt; s_s += s;
  }
  __shared__ float sm0[256], sm1[256];
  sm0[t] = s_sw; sm1[t] = s_s;
  __syncthreads();
  for (int off = 128; off > 0; off >>= 1) {
    if (t < off) { sm0[t] += sm0[t + off]; sm1[t] += sm1[t + off]; }
    __syncthreads();
  }
  if (t == 0) {
    partials[blockIdx.x * 2 + 0] = sm0[0];
    partials[blockIdx.x * 2 + 1] = sm1[0];
  }
}

// Single-block, fixed-order accumulate of partials into acc (deterministic; no float atomics)
__global__ __launch_bounds__(256) void k_acc_accum(const float* __restrict__ partials,
                                                   int nblocks, int nsums,
                                                   float* __restrict__ accdst) {
  __shared__ float sm[256];
  const int t = threadIdx.x;
  for (int j = 0; j < nsums; ++j) {
    float s = 0.0f;
    for (int i = t; i < nblocks; i += 256) s += partials[i * nsums + j];
    sm[t] = s;
    __syncthreads();
    for (int off = 128; off > 0; off >>= 1) {
      if (t < off) sm[t] += sm[t + off];
      __syncthreads();
    }
    if (t == 0) accdst[j] += sm[0];
    __syncthreads();
  }
}

__global__ void k_finalize(const float* __restrict__ acc, float* __restrict__ out) {
  if (threadIdx.x == 0 && blockIdx.x == 0) {
    const float sg = acc[0], sp = acc[1], spg = acc[2];
    const float a1 = acc[3], a2 = acc[4], b1 = acc[5], b2 = acc[6];
    const float dice = 1.0f - (2.0f * spg + 1.0f) / (sg + sp + 1.0f);
    const float tprec = (a1 + 1.0f) / (a2 + 1.0f);
    const float tsens = (b1 + 1.0f) / (b2 + 1.0f);
    const float cld = 1.0f - 2.0f * (tprec * tsens) / (tprec + tsens);
    out[0] = dice + cld;
  }
}

extern "C" void kernel_launch(void* const* d_in, const int* in_sizes, int n_in,
                              void* d_out, int out_size, void* d_ws, size_t ws_size,
                              hipStream_t stream) {
  const float* yp = (const float*)d_in[0];
  const int*   yt = (const int*)d_in[1];
  float* out = (float*)d_out;
  float* ws  = (float*)d_ws;

  float* acc      = ws;                          // 16 floats
  float* partials = ws + 16;                     // RED_BLOCKS*3 floats
  float* bufA     = ws + 16 + RED_BLOCKS * 3;    // VOL
  float* bufB     = bufA + VOL;                  // VOL
  float* skel     = bufB + VOL;                  // VOL

  const int EB = (VOL / 4) / 256;                // 3456 blocks, exact
  const dim3 sgrid(WW / 32, HH / 8, DD / 24);    // 6 x 24 x 4 = 576 blocks

  k_zero_acc<<<1, 16, 0, stream>>>(acc);
  k_reduce_dice<<<RED_BLOCKS, 256, 0, stream>>>(yp, yt, partials);
  k_acc_accum<<<1, 256, 0, stream>>>(partials, RED_BLOCKS, 3, acc + 0);

  for (int b = 0; b < 2; ++b) {
    for (int mode = 0; mode < 2; ++mode) {       // 0: skel of p1, 1: skel of one-hot
      k_init_field<<<EB, 256, 0, stream>>>(bufA, yp, yt, b, mode);
      float* prev = bufA;
      float* cur  = bufB;
      for (int k = 0; k < 4; ++k) {              // fused erode + dilate + skel update
        k_stage<<<sgrid, 256, 0, stream>>>(prev, cur, skel,
                                           (k == 0) ? 1 : 0,   // first
                                           (k < 3) ? 1 : 0);   // last stage: skip cur store
        float* tswap = prev; prev = cur; cur = tswap;
      }
      k_reduce_skel<<<RED_BLOCKS, 256, 0, stream>>>(skel, yp, yt, b, mode, partials);
      k_acc_accum<<<1, 256, 0, stream>>>(partials, RED_BLOCKS, 2, acc + 3 + mode * 2);
    }
  }
  k_finalize<<<1, 1, 0, stream>>>(acc, out);

  (void)in_sizes; (void)n_in; (void)out_size; (void)ws_size;
}